// IndividualGraphModuleConcat_53034256171384
// MI455X (gfx1250) — compile-verified
//
#include <hip/hip_runtime.h>
#include <math.h>

typedef __attribute__((ext_vector_type(2))) float v2f;
typedef __attribute__((ext_vector_type(8))) float v8f;

#define B_DIM 2
#define C_DIM 256
#define N_DIM 384   // T*P = 32*12

// ---------------------------------------------------------------------------
// Hardware tanh: CDNA5 has V_TANH_F32 as a TRANS op (co-executes with VALU).
// Prefer the clang builtin if declared; otherwise emit the instruction
// directly. OCML tanhf would be ~15 VALU ops; this is 1 TRANS op.
// ---------------------------------------------------------------------------
static __device__ __forceinline__ float hw_tanhf(float x) {
#if __has_builtin(__builtin_amdgcn_tanhf)
    return __builtin_amdgcn_tanhf(x);
#else
    float r;
    asm("v_tanh_f32 %0, %1" : "=v"(r) : "v"(x));
    return r;
#endif
}

// ---------------------------------------------------------------------------
// Small kernel: c1 = W1 @ theta_b, c2 = W2 @ phi_b  (256-long dots, 256 thr)
// ---------------------------------------------------------------------------
__global__ void fuse_bias_kernel(const float* __restrict__ W_a,
                                 const float* __restrict__ theta_b,
                                 const float* __restrict__ phi_b,
                                 float* __restrict__ c1,
                                 float* __restrict__ c2)
{
    int o2 = blockIdx.x * blockDim.x + threadIdx.x;
    if (o2 >= C_DIM) return;
    float s1 = 0.f, s2 = 0.f;
    const float* row = W_a + (long)o2 * (2 * C_DIM);
    #pragma unroll 8
    for (int o = 0; o < C_DIM; ++o) {
        s1 += row[o] * theta_b[o];
        s2 += row[C_DIM + o] * phi_b[o];
    }
    c1[o2] = s1;
    c2[o2] = s2;
}

// ---------------------------------------------------------------------------
// f32 WMMA GEMM:  D[M x N] = A[M x K] @ B[K x N] (+ bias[row])
// Row-major, arbitrary leading dims, batch over blockIdx.z via strides on B/D.
// One wave computes one 16x16 D tile using V_WMMA_F32_16X16X4_F32 (K-step 4).
//
// ISA VGPR layouts (CDNA5 ISA):
//   A 16x4 : lane(l&15)=M row; half=l>>4; vgpr j holds K = 2*half + j
//   B 4x16 : lane(l&15)=N col; vgpr j holds K = 2*half + j
//   C/D    : vgpr v holds row M = v + 8*half, col = l&15
// ---------------------------------------------------------------------------
__global__ void wmma_gemm_f32(const float* __restrict__ A, int lda,
                              const float* __restrict__ Bm, int ldb, long strideB,
                              float* __restrict__ D, int ldd, long strideD,
                              const float* __restrict__ bias,
                              int M, int N, int K)
{
    const float* Bp = Bm + (long)blockIdx.z * strideB;
    float* Dp = D + (long)blockIdx.z * strideD;

    const int wavesPerBlock = blockDim.x >> 5;
    const int wave = blockIdx.x * wavesPerBlock + (threadIdx.x >> 5);
    const int lane = threadIdx.x & 31;
    const int tilesN = N >> 4;
    const int tM = wave / tilesN;
    const int tN = wave - tM * tilesN;
    if (tM * 16 >= M) return;                 // wave-uniform guard

    const int half = lane >> 4;               // 0 or 1
    const int ln   = lane & 15;
    const int rowA = tM * 16 + ln;            // A row held by this lane
    const int colB = tN * 16 + ln;            // B col held by this lane

    v8f acc = {};
    #pragma unroll 8
    for (int k0 = 0; k0 < K; k0 += 4) {
        const int ka = k0 + 2 * half;         // K index for vgpr 0 (vgpr1 = ka+1)
        v2f af = *(const v2f*)(A + (long)rowA * lda + ka);   // ka even -> 8B aligned
        v2f bf;
        bf.x = Bp[(long)ka * ldb + colB];
        bf.y = Bp[(long)(ka + 1) * ldb + colB];
        acc = __builtin_amdgcn_wmma_f32_16x16x4_f32(
            /*neg_a=*/false, af, /*neg_b=*/false, bf,
            /*c_mod=*/(short)0, acc, /*reuse_a=*/false, /*reuse_b=*/false);
    }

    #pragma unroll
    for (int v = 0; v < 8; ++v) {
        const int row = tM * 16 + v + 8 * half;
        float val = acc[v];
        if (bias) val += bias[row];
        Dp[(long)row * ldd + tN * 16 + ln] = val;
    }
}

// ---------------------------------------------------------------------------
// Pairwise interaction:  f[b,i,k] = sum_o v[o] * tanh(a[b,o,k] + bt[b,o,i])
// Grid: (N/32 k-tiles, N/32 i-tiles, B). 256 threads; o staged in 64-row
// LDS chunks. Thread t: k = t&31 (conflict-free), i = (t>>5) + {0,8,16,24}
// (wave-broadcast reads of bt_s). Inner body: 1 v_add + 1 v_tanh_f32 (TRANS,
// co-executes) + 1 v_fma per element.
// ---------------------------------------------------------------------------
#define OCH 64
__global__ void pairwise_tanh_kernel(const float* __restrict__ a,
                                     const float* __restrict__ bt,
                                     const float* __restrict__ v,
                                     float* __restrict__ f)
{
    __shared__ float a_s[OCH][33];
    __shared__ float bt_s[OCH][33];
    __shared__ float v_s[OCH];

    const int b     = blockIdx.z;
    const int kBase = blockIdx.x * 32;
    const int iBase = blockIdx.y * 32;
    const float* ab  = a  + (long)b * C_DIM * N_DIM;
    const float* btb = bt + (long)b * C_DIM * N_DIM;

    const int t  = threadIdx.x;
    const int kl = t & 31;
    const int ig = t >> 5;        // 0..7

    float acc0 = 0.f, acc1 = 0.f, acc2 = 0.f, acc3 = 0.f;

    for (int oc = 0; oc < C_DIM; oc += OCH) {
        for (int r = ig; r < OCH; r += 8) {
            a_s[r][kl]  = ab [(long)(oc + r) * N_DIM + kBase + kl];
            bt_s[r][kl] = btb[(long)(oc + r) * N_DIM + iBase + kl];
        }
        if (t < OCH) v_s[t] = v[oc + t];
        __syncthreads();

        #pragma unroll 4
        for (int o = 0; o < OCH; ++o) {
            const float av = a_s[o][kl];
            const float vo = v_s[o];
            acc0 += vo * hw_tanhf(av + bt_s[o][ig]);
            acc1 += vo * hw_tanhf(av + bt_s[o][ig + 8]);
            acc2 += vo * hw_tanhf(av + bt_s[o][ig + 16]);
            acc3 += vo * hw_tanhf(av + bt_s[o][ig + 24]);
        }
        __syncthreads();
    }

    float* fb = f + (long)b * N_DIM * N_DIM;
    fb[(long)(iBase + ig     ) * N_DIM + kBase + kl] = acc0;
    fb[(long)(iBase + ig +  8) * N_DIM + kBase + kl] = acc1;
    fb[(long)(iBase + ig + 16) * N_DIM + kBase + kl] = acc2;
    fb[(long)(iBase + ig + 24) * N_DIM + kBase + kl] = acc3;
}

// ---------------------------------------------------------------------------
// Softmax over the i axis (rows) per (b, k) column. Block = (32 k, 8 i-rows).
// k-contiguous global accesses are coalesced; i-partials reduced in LDS.
// Inputs are max-subtracted -> __expf (native v_exp_f32) is range-safe.
// ---------------------------------------------------------------------------
__global__ void softmax_i_kernel(const float* __restrict__ f,
                                 float* __restrict__ out)
{
    __shared__ float red[8][33];
    __shared__ float colM[32];
    __shared__ float colS[32];

    const int b     = blockIdx.y;
    const int kBase = blockIdx.x * 32;
    const int kl    = threadIdx.x & 31;
    const int iy    = threadIdx.x >> 5;     // 0..7
    const float* fb = f + (long)b * N_DIM * N_DIM;
    float* ob       = out + (long)b * N_DIM * N_DIM;

    // 1) column max
    float m = -INFINITY;
    for (int i = iy; i < N_DIM; i += 8)
        m = fmaxf(m, fb[(long)i * N_DIM + kBase + kl]);
    red[iy][kl] = m;
    __syncthreads();
    if (iy == 0) {
        float mm = red[0][kl];
        #pragma unroll
        for (int r = 1; r < 8; ++r) mm = fmaxf(mm, red[r][kl]);
        colM[kl] = mm;
    }
    __syncthreads();
    const float mm = colM[kl];

    // 2) column sum of exp
    float s = 0.f;
    for (int i = iy; i < N_DIM; i += 8)
        s += __expf(fb[(long)i * N_DIM + kBase + kl] - mm);
    red[iy][kl] = s;
    __syncthreads();
    if (iy == 0) {
        float ss = 0.f;
        #pragma unroll
        for (int r = 0; r < 8; ++r) ss += red[r][kl];
        colS[kl] = ss;
    }
    __syncthreads();
    const float inv = 1.f / colS[kl];

    // 3) write normalized output
    for (int i = iy; i < N_DIM; i += 8) {
        const long idx = (long)i * N_DIM + kBase + kl;
        ob[idx] = __expf(fb[idx] - mm) * inv;
    }
}

// ---------------------------------------------------------------------------
// Host-side launch
// ---------------------------------------------------------------------------
extern "C" void kernel_launch(void* const* d_in, const int* in_sizes, int n_in,
                              void* d_out, int out_size, void* d_ws, size_t ws_size,
                              hipStream_t stream)
{
    const float* x       = (const float*)d_in[0]; // (B,C,T,P) == (B,C,N) contiguous
    const float* theta_w = (const float*)d_in[1]; // (C,C)
    const float* theta_b = (const float*)d_in[2]; // (C,)
    const float* phi_w   = (const float*)d_in[3]; // (C,C)
    const float* phi_b   = (const float*)d_in[4]; // (C,)
    const float* W_a     = (const float*)d_in[5]; // (C,2C)
    const float* v_a     = (const float*)d_in[6]; // (1,C)
    float* out = (float*)d_out;                   // (B,N,N)

    // Workspace layout (floats)
    float* ws = (float*)d_ws;
    float* M1 = ws;                         // 256*256
    float* M2 = M1 + C_DIM * C_DIM;         // 256*256
    float* c1 = M2 + C_DIM * C_DIM;         // 256
    float* c2 = c1 + C_DIM;                 // 256
    float* a  = c2 + C_DIM;                 // B*C*N
    float* bt = a  + B_DIM * C_DIM * N_DIM; // B*C*N
    float* f  = bt + B_DIM * C_DIM * N_DIM; // B*N*N

    // 1) fused bias vectors
    fuse_bias_kernel<<<1, 256, 0, stream>>>(W_a, theta_b, phi_b, c1, c2);

    // 2) fused weight matrices: M1 = W1 @ theta_w, M2 = W2 @ phi_w
    //    W1 rows have stride 2C inside W_a; W2 starts at column C.
    {
        // tiles: (256/16)*(256/16) = 256 waves -> 32 blocks of 8 waves
        dim3 grid(32, 1, 1), block(256, 1, 1);
        wmma_gemm_f32<<<grid, block, 0, stream>>>(
            W_a, 2 * C_DIM, theta_w, C_DIM, 0, M1, C_DIM, 0,
            nullptr, C_DIM, C_DIM, C_DIM);
        wmma_gemm_f32<<<grid, block, 0, stream>>>(
            W_a + C_DIM, 2 * C_DIM, phi_w, C_DIM, 0, M2, C_DIM, 0,
            nullptr, C_DIM, C_DIM, C_DIM);
    }

    // 3) batched activation GEMMs: a = M1 @ x + c1 ; bt = M2 @ x + c2
    {
        // tiles: (256/16)*(384/16) = 384 waves -> 48 blocks of 8 waves, z = B
        dim3 grid(48, 1, B_DIM), block(256, 1, 1);
        const long sAct = (long)C_DIM * N_DIM;
        wmma_gemm_f32<<<grid, block, 0, stream>>>(
            M1, C_DIM, x, N_DIM, sAct, a, N_DIM, sAct,
            c1, C_DIM, N_DIM, C_DIM);
        wmma_gemm_f32<<<grid, block, 0, stream>>>(
            M2, C_DIM, x, N_DIM, sAct, bt, N_DIM, sAct,
            c2, C_DIM, N_DIM, C_DIM);
    }

    // 4) pairwise tanh interaction -> f
    {
        dim3 grid(N_DIM / 32, N_DIM / 32, B_DIM), block(256, 1, 1);
        pairwise_tanh_kernel<<<grid, block, 0, stream>>>(a, bt, v_a, f);
    }

    // 5) softmax over i
    {
        dim3 grid(N_DIM / 32, B_DIM, 1), block(256, 1, 1);
        softmax_i_kernel<<<grid, block, 0, stream>>>(f, out);
    }
}